// ArithmeticNps_88940182766237
// MI455X (gfx1250) — compile-verified
//
#include <hip/hip_runtime.h>
#include <hip/hip_bf16.h>

typedef float v2f __attribute__((ext_vector_type(2)));
typedef float v8f __attribute__((ext_vector_type(8)));

#define CV 128
#define CM 128
#define DPC 256   // 2*CV
#define NR 16

// ---------------------------------------------------------------------------
// Kernel 0: precompute read1 (16x32) = einsum('nr,nrm->nm', rules_emb, s1_k_w)
//           + s1_k_b ; also zero bin counters.
// ---------------------------------------------------------------------------
__global__ void read1_kernel(const float* __restrict__ rules_emb,
                             const float* __restrict__ s1_k_w,
                             const float* __restrict__ s1_k_b,
                             float* __restrict__ read1,
                             int* __restrict__ binCount) {
    int t = threadIdx.x;              // 512 threads
    if (t < NR) binCount[t] = 0;
    int n = t >> 5, m = t & 31;
    float s = s1_k_b[n * 32 + m];
    for (int r = 0; r < 64; ++r)
        s += rules_emb[n * 64 + r] * s1_k_w[(n * 64 + r) * 32 + m];
    read1[t] = s;
}

// ---------------------------------------------------------------------------
// Kernel 1: per-element front end. One 128-thread block per batch element.
// Produces var_pc (B x 256), idx_r (B), and rule histogram.
// ---------------------------------------------------------------------------
__global__ __launch_bounds__(128) void phase1_kernel(
    const float* __restrict__ op1, const float* __restrict__ op2,
    const int* __restrict__ oper,
    const float* __restrict__ ew1, const float* __restrict__ eb1,
    const float* __restrict__ ew2, const float* __restrict__ eb2,
    const float* __restrict__ ow1, const float* __restrict__ ob1,
    const float* __restrict__ ow2, const float* __restrict__ ob2,
    const float* __restrict__ rules_emb,
    const float* __restrict__ s1_q_w, const float* __restrict__ s1_q_b,
    const float* __restrict__ read1,
    const float* __restrict__ s2_q_w, const float* __restrict__ s2_q_b,
    const float* __restrict__ s2_k_w, const float* __restrict__ s2_k_b,
    float* __restrict__ varpc, int* __restrict__ idxR,
    int* __restrict__ binCount) {
    __shared__ float h1a[64], h1b[64], hop[64];
    __shared__ float hidden[3][CV];
    __shared__ float w1s[3][32];
    __shared__ float att1[48];
    __shared__ float r2[2][16], w2s2[2][16];
    __shared__ int sidx[3];

    int e = blockIdx.x;
    int t = threadIdx.x;
    float v1 = op1[e], v2 = op2[e];
    int op = oper[e];

    // first encoder layer (x1c=[v1,0], x2c=[v2,1], opc=onehot(op))
    if (t < 64) {
        float pa = v1 * ew1[t] + eb1[t];
        h1a[t] = pa > 0.f ? pa : 0.f;
        float pb = v2 * ew1[t] + ew1[64 + t] + eb1[t];
        h1b[t] = pb > 0.f ? pb : 0.f;
        float po = ow1[op * 64 + t] + ob1[t];
        hop[t] = po > 0.f ? po : 0.f;
    }
    __syncthreads();

    // second encoder layer -> hidden (3 x 128), thread t owns channel t
    {
        float a = eb2[t], b = eb2[t], c = ob2[t];
        for (int j = 0; j < 64; ++j) {
            float wa = ew2[j * CV + t];
            a += h1a[j] * wa;
            b += h1b[j] * wa;
            c += hop[j] * ow2[j * CV + t];
        }
        hidden[0][t] = a; hidden[1][t] = b; hidden[2][t] = c;
    }
    __syncthreads();

    // write1 = hidden @ s1_q_w + b   (3 x 32)
    if (t < 96) {
        int k = t >> 5, m = t & 31;
        float s = s1_q_b[m];
        for (int c = 0; c < CV; ++c) s += hidden[k][c] * s1_q_w[c * 32 + m];
        w1s[k][m] = s;
    }
    __syncthreads();

    // attent1[n,k]  (scale dropped: positive constant, argmax-invariant)
    if (t < 48) {
        int n = t / 3, k = t - n * 3;
        float s = 0.f;
        for (int m = 0; m < 32; ++m) s += read1[n * 32 + m] * w1s[k][m];
        att1[t] = s;
    }
    __syncthreads();
    if (t == 0) {  // first-max argmax over flattened (n*3+k)
        int best = 0; float bv = att1[0];
        for (int i = 1; i < 48; ++i)
            if (att1[i] > bv) { bv = att1[i]; best = i; }
        sidx[0] = best / 3;
    }
    __syncthreads();
    int idx_r = sidx[0];
    const float* rb = rules_emb + idx_r * 64;

    // read2 (2x16) from rule body; write2 (2x16) from hidden2
    if (t < 32) {
        int n = t >> 4, m = t & 15;
        float s = s2_k_b[n * 16 + m];
        for (int r = 0; r < 64; ++r) s += rb[r] * s2_k_w[(n * 64 + r) * 16 + m];
        r2[n][m] = s;
        float q = s2_q_b[n * 16 + m];
        for (int c = 0; c < CV; ++c) q += hidden[n][c] * s2_q_w[(n * CV + c) * 16 + m];
        w2s2[n][m] = q;
    }
    __syncthreads();
    if (t == 0) {
        float a00 = 0.f, a01 = 0.f, a10 = 0.f, a11 = 0.f;
        for (int m = 0; m < 16; ++m) {
            a00 += r2[0][m] * w2s2[0][m];
            a01 += r2[0][m] * w2s2[1][m];
            a10 += r2[1][m] * w2s2[0][m];
            a11 += r2[1][m] * w2s2[1][m];
        }
        sidx[1] = (a01 > a00) ? 1 : 0;   // idx_p (first-max tie break)
        sidx[2] = (a11 > a10) ? 1 : 0;   // idx_c
    }
    __syncthreads();
    int ip = sidx[1], ic = sidx[2];
    varpc[(size_t)e * DPC + t] = hidden[ip][t];
    varpc[(size_t)e * DPC + CV + t] = hidden[ic][t];
    if (t == 0) {
        idxR[e] = idx_r;
        atomicAdd(&binCount[idx_r], 1);
    }
}

// ---------------------------------------------------------------------------
// Kernel 2: exclusive scan of 16 bins
// ---------------------------------------------------------------------------
__global__ void scan_kernel(const int* __restrict__ binCount,
                            int* __restrict__ binStart,
                            int* __restrict__ binCursor) {
    if (threadIdx.x == 0) {
        int acc = 0;
        for (int i = 0; i < NR; ++i) {
            binStart[i] = acc; binCursor[i] = acc; acc += binCount[i];
        }
        binStart[NR] = acc;
    }
}

// ---------------------------------------------------------------------------
// Kernel 3: scatter element indices into rule-sorted order
// ---------------------------------------------------------------------------
__global__ void scatter_kernel(const int* __restrict__ idxR,
                               int* __restrict__ binCursor,
                               int* __restrict__ perm, int B) {
    int i = blockIdx.x * blockDim.x + threadIdx.x;
    if (i < B) {
        int pos = atomicAdd(&binCursor[idxR[i]], 1);
        perm[pos] = i;
    }
}

// ---------------------------------------------------------------------------
// Kernel 4: per-rule GEMMs via V_WMMA_F32_16X16X4_F32 + fused decoder.
// grid = (B/16 tiles, 16 rules), one wave32 per tile of 16 elements.
// ---------------------------------------------------------------------------
__global__ __launch_bounds__(32) void rule_gemm_kernel(
    const float* __restrict__ varpc, const int* __restrict__ perm,
    const int* __restrict__ binStart,
    const float* __restrict__ rule_W1, const float* __restrict__ rule_b1,
    const float* __restrict__ rule_W2, const float* __restrict__ rule_b2,
    const float* __restrict__ dec_w1, const float* __restrict__ dec_b1,
    const float* __restrict__ dec_w2, const float* __restrict__ dec_b2,
    float* __restrict__ out) {
    __shared__ float X[16][DPC];   // input tile, later reused for O (16x128)
    __shared__ float H[16][CM];    // hidden tile
    __shared__ int rows[16];

    int r = blockIdx.y;
    int tile = blockIdx.x;
    int start = binStart[r];
    int cnt = binStart[r + 1] - start;
    if (tile * 16 >= cnt) return;              // uniform per wave

    int lane = threadIdx.x;
    int nvalid = cnt - tile * 16; if (nvalid > 16) nvalid = 16;

    if (lane < 16) {
        int idx = tile * 16 + lane;
        if (idx >= cnt) idx = cnt - 1;         // pad with last valid row
        rows[lane] = perm[start + idx];
    }
    __syncthreads();

    // stage X tile (gather var_pc rows)
    for (int i = lane; i < 16 * DPC; i += 32) {
        int m = i >> 8, c = i & (DPC - 1);
        X[m][c] = varpc[(size_t)rows[m] * DPC + c];
    }
    __syncthreads();

    const float* W1r = rule_W1 + (size_t)r * DPC * CM;
    const float* W2r = rule_W2 + (size_t)r * CM * CV;
    const float* b1r = rule_b1 + r * CM;
    const float* b2r = rule_b2 + r * CV;

    int half = lane >> 4;        // ISA A-layout: lanes16-31 hold K+2,K+3
    int l16  = lane & 15;

    // ---- GEMM1: H = relu(X(16x256) @ W1r(256x128) + b1) ----
    for (int n = 0; n < CM / 16; ++n) {
        v8f acc = {};
        int ncol = n * 16 + l16;
        for (int k0 = 0; k0 < DPC; k0 += 4) {
            v2f a, b;
            a.x = X[l16][k0 + 2 * half];
            a.y = X[l16][k0 + 2 * half + 1];
            b.x = W1r[(size_t)(k0 + 2 * half) * CM + ncol];
            b.y = W1r[(size_t)(k0 + 2 * half + 1) * CM + ncol];
            acc = __builtin_amdgcn_wmma_f32_16x16x4_f32(
                false, a, false, b, (short)0, acc, false, false);
        }
        float bias = b1r[ncol];
        #pragma unroll
        for (int i = 0; i < 8; ++i) {
            float v = acc[i] + bias;                 // row M = i + 8*half
            H[i + 8 * half][ncol] = v > 0.f ? v : 0.f;
        }
    }
    __syncthreads();

    // ---- GEMM2: O = H(16x128) @ W2r(128x128) + b2  (store into X) ----
    for (int n = 0; n < CV / 16; ++n) {
        v8f acc = {};
        int ncol = n * 16 + l16;
        for (int k0 = 0; k0 < CM; k0 += 4) {
            v2f a, b;
            a.x = H[l16][k0 + 2 * half];
            a.y = H[l16][k0 + 2 * half + 1];
            b.x = W2r[(size_t)(k0 + 2 * half) * CV + ncol];
            b.y = W2r[(size_t)(k0 + 2 * half + 1) * CV + ncol];
            acc = __builtin_amdgcn_wmma_f32_16x16x4_f32(
                false, a, false, b, (short)0, acc, false, false);
        }
        float bias = b2r[ncol];
        #pragma unroll
        for (int i = 0; i < 8; ++i)
            X[i + 8 * half][ncol] = acc[i] + bias;
    }
    __syncthreads();

    // ---- fused decoder: out = relu(O @ dec_w1 + db1) @ dec_w2 + db2 ----
    if (lane < 16 && lane < nvalid) {
        float accum = dec_b2[0];
        for (int j = 0; j < 64; ++j) {
            float pre = dec_b1[j];
            for (int c = 0; c < CV; ++c) pre += X[lane][c] * dec_w1[c * 64 + j];
            accum += (pre > 0.f ? pre : 0.f) * dec_w2[j];
        }
        out[rows[lane]] = accum;
    }
}

// ---------------------------------------------------------------------------
extern "C" void kernel_launch(void* const* d_in, const int* in_sizes, int n_in,
                              void* d_out, int out_size, void* d_ws, size_t ws_size,
                              hipStream_t stream) {
    const int B = in_sizes[0];

    const float* op1      = (const float*)d_in[0];
    const float* op2      = (const float*)d_in[1];
    const int*   oper     = (const int*)  d_in[2];
    const float* ew1      = (const float*)d_in[3];
    const float* eb1      = (const float*)d_in[4];
    const float* ew2      = (const float*)d_in[5];
    const float* eb2      = (const float*)d_in[6];
    const float* ow1      = (const float*)d_in[7];
    const float* ob1      = (const float*)d_in[8];
    const float* ow2      = (const float*)d_in[9];
    const float* ob2      = (const float*)d_in[10];
    const float* dec_w1   = (const float*)d_in[11];
    const float* dec_b1   = (const float*)d_in[12];
    const float* dec_w2   = (const float*)d_in[13];
    const float* dec_b2   = (const float*)d_in[14];
    const float* rules    = (const float*)d_in[15];
    const float* rule_W1  = (const float*)d_in[16];
    const float* rule_b1  = (const float*)d_in[17];
    const float* rule_W2  = (const float*)d_in[18];
    const float* rule_b2  = (const float*)d_in[19];
    const float* s1_q_w   = (const float*)d_in[20];
    const float* s1_q_b   = (const float*)d_in[21];
    const float* s1_k_w   = (const float*)d_in[22];
    const float* s1_k_b   = (const float*)d_in[23];
    const float* s2_q_w   = (const float*)d_in[24];
    const float* s2_q_b   = (const float*)d_in[25];
    const float* s2_k_w   = (const float*)d_in[26];
    const float* s2_k_b   = (const float*)d_in[27];

    char* ws = (char*)d_ws;
    size_t o = 0;
    float* varpc     = (float*)(ws + o); o += (size_t)B * DPC * sizeof(float);
    float* read1     = (float*)(ws + o); o += 512 * sizeof(float);
    int*   idxR      = (int*)  (ws + o); o += (size_t)B * sizeof(int);
    int*   binCount  = (int*)  (ws + o); o += 64;
    int*   binStart  = (int*)  (ws + o); o += 128;   // 17 ints + pad
    int*   binCursor = (int*)  (ws + o); o += 64;
    int*   perm      = (int*)  (ws + o); o += (size_t)B * sizeof(int);
    (void)ws_size; (void)n_in; (void)out_size;

    read1_kernel<<<1, 512, 0, stream>>>(rules, s1_k_w, s1_k_b, read1, binCount);

    phase1_kernel<<<B, 128, 0, stream>>>(
        op1, op2, oper, ew1, eb1, ew2, eb2, ow1, ob1, ow2, ob2, rules,
        s1_q_w, s1_q_b, read1, s2_q_w, s2_q_b, s2_k_w, s2_k_b,
        varpc, idxR, binCount);

    scan_kernel<<<1, 32, 0, stream>>>(binCount, binStart, binCursor);

    scatter_kernel<<<(B + 255) / 256, 256, 0, stream>>>(idxR, binCursor, perm, B);

    dim3 grid(B / 16, NR);
    rule_gemm_kernel<<<grid, 32, 0, stream>>>(
        varpc, perm, binStart, rule_W1, rule_b1, rule_W2, rule_b2,
        dec_w1, dec_b1, dec_w2, dec_b2, (float*)d_out);
}